// WeightedCrossEntropyLoss_16260746182668
// MI455X (gfx1250) — compile-verified
//
#include <hip/hip_runtime.h>

#define NUM_CLASSES 19
#define BATCH 8
#define HH 512
#define WW 1024
#define HW (HH * WW)          /* 524288 = 2^19 */
#define NPIX (BATCH * HW)     /* 4194304 */
#define NB 1024               /* blocks in loss kernel == #partial pairs */
#define NT 256

typedef __attribute__((ext_vector_type(2))) float v2f;
typedef __attribute__((ext_vector_type(8))) float v8f;

// ---------------------------------------------------------------- init ----
__global__ void wce_init(unsigned int* __restrict__ counts) {
    if (threadIdx.x < NUM_CLASSES) counts[threadIdx.x] = 0u;
}

// ----------------------------------------------------------- histogram ----
__global__ void wce_hist(const int* __restrict__ tgt,
                         unsigned int* __restrict__ counts) {
    __shared__ unsigned int lbin[NUM_CLASSES];
    if (threadIdx.x < NUM_CLASSES) lbin[threadIdx.x] = 0u;
    __syncthreads();
    const int stride = gridDim.x * blockDim.x;
    for (int i = blockIdx.x * blockDim.x + threadIdx.x; i < NPIX; i += stride) {
        const int t = tgt[i];                 // RT loads: warm L2 for loss pass
        if (t >= 0 && t < NUM_CLASSES) atomicAdd(&lbin[t], 1u);
    }
    __syncthreads();
    if (threadIdx.x < NUM_CLASSES) atomicAdd(&counts[threadIdx.x], lbin[threadIdx.x]);
}

// ---------------------------------------------------- streaming loss ------
// One pixel per thread per iteration; lanes take consecutive pixels so each
// of the 19 per-class loads is a fully coalesced 128B/wave transaction.
// Logits are non-temporal (319MB stream > 192MB L2).
__global__ void wce_loss(const float* __restrict__ logits,
                         const int* __restrict__ tgt,
                         const unsigned int* __restrict__ counts,
                         float* __restrict__ partials) {
    __shared__ float wcls[NUM_CLASSES];
    __shared__ float sn[8], sd[8];
    if (threadIdx.x < NUM_CLASSES)
        wcls[threadIdx.x] = 1.0f - (float)counts[threadIdx.x] * (1.0f / (float)NPIX);
    __syncthreads();

    float num = 0.0f, den = 0.0f;
    const int stride = gridDim.x * blockDim.x;
    for (int p = blockIdx.x * blockDim.x + threadIdx.x; p < NPIX; p += stride) {
        const int b  = p >> 19;           // p / HW
        const int hw = p & (HW - 1);      // p % HW
        const float* base = logits + (size_t)b * NUM_CLASSES * HW + hw;
        const int t = tgt[p];

        float x[NUM_CLASSES];
        float m = -3.4e38f;
#pragma unroll
        for (int c = 0; c < NUM_CLASSES; ++c) {
            x[c] = __builtin_nontemporal_load(base + (size_t)c * HW);
            m = fmaxf(m, x[c]);
        }
        float s  = 0.0f;
        float xt = 0.0f;
#pragma unroll
        for (int c = 0; c < NUM_CLASSES; ++c) {
            s += __expf(x[c] - m);
            xt = (c == t) ? x[c] : xt;    // compile-time c: stays in VGPRs
        }
        const float lse = m + __logf(s);
        if (t >= 0) {                     // ignore_index handling
            const float w = wcls[t];
            num = fmaf(w, lse - xt, num); // -w*log p_t
            den += w;
        }
    }

    // wave32 reduction
#pragma unroll
    for (int off = 16; off > 0; off >>= 1) {
        num += __shfl_xor(num, off, 32);
        den += __shfl_xor(den, off, 32);
    }
    const int wid = threadIdx.x >> 5;
    if ((threadIdx.x & 31) == 0) { sn[wid] = num; sd[wid] = den; }
    __syncthreads();
    if (threadIdx.x == 0) {
        float a = 0.0f, d = 0.0f;
#pragma unroll
        for (int i = 0; i < 8; ++i) { a += sn[i]; d += sd[i]; }
        partials[2 * blockIdx.x + 0] = a;
        partials[2 * blockIdx.x + 1] = d;
    }
}

// -------------------------------- WMMA tree reduction of block partials ---
// One wave (EXEC all-ones, as WMMA requires). Per iteration each lane loads
// 2 f32 partials into A(16x4); B = ones => D[i][j] = rowsum_i + C[i][j].
// After the loop, column 0 of C (lane0 v[0..7] = rows 0..7, lane16 v[0..7]
// = rows 8..15) sums to the grand total.
__global__ void wce_finalize(const float* __restrict__ partials,
                             float* __restrict__ out) {
    const int lane = threadIdx.x;         // 0..31, blockDim.x == 32
    v8f cn = {};
    v8f cd = {};
    v2f ones; ones[0] = 1.0f; ones[1] = 1.0f;

    for (int base = 0; base < NB; base += 64) {
        const int i0 = base + 2 * lane;   // this lane's two block-indices
        v2f an, ad;
        an[0] = partials[2 * i0 + 0];
        an[1] = partials[2 * i0 + 2];
        ad[0] = partials[2 * i0 + 1];
        ad[1] = partials[2 * i0 + 3];
        cn = __builtin_amdgcn_wmma_f32_16x16x4_f32(false, an, false, ones,
                                                   (short)0, cn, false, false);
        cd = __builtin_amdgcn_wmma_f32_16x16x4_f32(false, ad, false, ones,
                                                   (short)0, cd, false, false);
    }

    float s_n = cn[0] + cn[1] + cn[2] + cn[3] + cn[4] + cn[5] + cn[6] + cn[7];
    float s_d = cd[0] + cd[1] + cd[2] + cd[3] + cd[4] + cd[5] + cd[6] + cd[7];
    const float tot_n = __shfl(s_n, 0, 32) + __shfl(s_n, 16, 32);
    const float tot_d = __shfl(s_d, 0, 32) + __shfl(s_d, 16, 32);
    if (lane == 0) out[0] = tot_n / tot_d;
}

// ------------------------------------------------------------------ API ---
extern "C" void kernel_launch(void* const* d_in, const int* in_sizes, int n_in,
                              void* d_out, int out_size, void* d_ws, size_t ws_size,
                              hipStream_t stream) {
    (void)in_sizes; (void)n_in; (void)out_size; (void)ws_size;
    const float* logits = (const float*)d_in[0];
    const int*   tgt    = (const int*)d_in[1];

    unsigned int* counts   = (unsigned int*)d_ws;              // 19 u32
    float*        partials = (float*)((char*)d_ws + 256);      // 2*NB f32
    float*        out      = (float*)d_out;

    wce_init<<<1, 32, 0, stream>>>(counts);
    wce_hist<<<512, NT, 0, stream>>>(tgt, counts);
    wce_loss<<<NB, NT, 0, stream>>>(logits, tgt, counts, partials);
    wce_finalize<<<1, 32, 0, stream>>>(partials, out);
}